// GradientProcessor_19258633356159
// MI455X (gfx1250) — compile-verified
//
#include <hip/hip_runtime.h>
#include <stdint.h>

// Problem constants (match reference)
#define BATCH 16
#define KBOX  16
#define HH    896
#define WW    896
#define CC    3
#define PP    64
#define NPATCH (BATCH * KBOX)        // 256
#define PIX    (PP * PP)             // 4096
#define OUT_N  (PP * PP * CC)        // 12288
#define IMG_STRIDE ((size_t)HH * WW * CC)
#define ROW_STRIDE (WW * CC)

typedef __attribute__((address_space(1))) int* gptr_i32;
typedef __attribute__((address_space(3))) int* lptr_i32;

// ---------------------------------------------------------------------------
// Bilinear coordinate setup, matching reference _coords() exactly:
//   pos = (p+0.5)*(h/P) - 0.5 ; clip [0, h-1] ; lo=floor ; hi=min(lo+1,h-1)
// ---------------------------------------------------------------------------
__device__ __forceinline__ void lerp_setup(int p, int sz, int start,
                                           int& i0, int& i1, float& w) {
  float s   = (float)sz;
  float pos = ((float)p + 0.5f) * (s * (1.0f / (float)PP)) - 0.5f;
  pos       = fminf(fmaxf(pos, 0.0f), s - 1.0f);
  float f   = floorf(pos);
  w         = pos - f;
  int lo    = (int)f;
  int hi    = (lo + 1 < sz) ? (lo + 1) : (sz - 1);
  i0 = start + lo;
  i1 = start + hi;
}

// ---------------------------------------------------------------------------
// Stage 1: each block = one 256-pixel tile x one slice of patches.
// Thread t owns pixel (tile*256 + t): accumulates 3 channels over the slice's
// patches in registers, writes partial[slice][pix*3 + c].
// Box descriptors for the slice are staged into LDS with the CDNA5 async
// global->LDS copy path when available.
// ---------------------------------------------------------------------------
__global__ __launch_bounds__(256) void gp_stage1(
    const float* __restrict__ grad,   // (B,H,W,C)
    const int*   __restrict__ boxes,  // (B,K,4) = ymin,xmin,h,w
    float*       __restrict__ partial,// (S, OUT_N)
    int patches_per_slice) {

  __shared__ int sbox[NPATCH * 4];    // worst case S=1: 4 KB

  const int slice      = blockIdx.y;
  const int base_patch = slice * patches_per_slice;
  const int n_ints     = patches_per_slice * 4;
  const int* gsrc      = boxes + base_patch * 4;

#if __has_builtin(__builtin_amdgcn_global_load_async_to_lds_b32)
  for (int i = threadIdx.x; i < n_ints; i += 256) {
    // generic->AS casts via integer (global flat addr == AS1 addr; LDS offset
    // == low 32 bits of flat shared address on AMDGPU)
    gptr_i32 g = (gptr_i32)(uintptr_t)(gsrc + i);
    lptr_i32 l = (lptr_i32)(uint32_t)(uintptr_t)(&sbox[i]);
    __builtin_amdgcn_global_load_async_to_lds_b32(g, l, 0, 0);
  }
#if __has_builtin(__builtin_amdgcn_s_wait_asynccnt)
  __builtin_amdgcn_s_wait_asynccnt(0);
#else
  asm volatile("s_wait_asynccnt 0" ::: "memory");
#endif
#else
  for (int i = threadIdx.x; i < n_ints; i += 256) sbox[i] = gsrc[i];
#endif
  __syncthreads();

  const int pix = blockIdx.x * 256 + threadIdx.x;  // 0..4095
  const int py  = pix >> 6;
  const int px  = pix & 63;

  float acc[3] = {0.0f, 0.0f, 0.0f};

  for (int j = 0; j < patches_per_slice; ++j) {
    const int ymin = sbox[j * 4 + 0];
    const int xmin = sbox[j * 4 + 1];
    const int h    = sbox[j * 4 + 2];
    const int w    = sbox[j * 4 + 3];
    const int b    = (base_patch + j) >> 4;   // K == 16
    const float* img = grad + (size_t)b * IMG_STRIDE;

    // Warm L2/WGP$ for the next patch's crop corner (gfx1250 global_prefetch)
    if (j + 1 < patches_per_slice) {
      const int nb = (base_patch + j + 1) >> 4;
      const int ny = sbox[(j + 1) * 4 + 0];
      const int nx = sbox[(j + 1) * 4 + 1];
      __builtin_prefetch(grad + (size_t)nb * IMG_STRIDE +
                             (size_t)ny * ROW_STRIDE + (size_t)nx * CC,
                         0, 1);
    }

    int y0, y1, x0, x1;
    float wy, wx;
    lerp_setup(py, h, ymin, y0, y1, wy);
    lerp_setup(px, w, xmin, x0, x1, wx);

    const float* r0 = img + (size_t)y0 * ROW_STRIDE;
    const float* r1 = img + (size_t)y1 * ROW_STRIDE;
    const float* tl = r0 + x0 * CC;
    const float* tr = r0 + x1 * CC;
    const float* bl = r1 + x0 * CC;
    const float* br = r1 + x1 * CC;

#pragma unroll
    for (int c = 0; c < CC; ++c) {
      float vtl = tl[c], vtr = tr[c], vbl = bl[c], vbr = br[c];
      float top = fmaf(wx, vtr - vtl, vtl);
      float bot = fmaf(wx, vbr - vbl, vbl);
      acc[c] += fmaf(wy, bot - top, top);
    }
  }

  float* dst = partial + (size_t)slice * OUT_N + (size_t)pix * CC;
  dst[0] = acc[0];
  dst[1] = acc[1];
  dst[2] = acc[2];
}

// ---------------------------------------------------------------------------
// Stage 2: fixed-order reduction over S slices + elementwise patch_grads.
// Safe when partial == out (each thread touches only its own index).
// ---------------------------------------------------------------------------
__global__ __launch_bounds__(256) void gp_stage2(
    const float* __restrict__ partial,
    const float* __restrict__ pgrads,
    float*       __restrict__ out,
    int nslices) {
  int i = blockIdx.x * 256 + threadIdx.x;
  if (i >= OUT_N) return;
  float s = 0.0f;
  for (int j = 0; j < nslices; ++j) s += partial[(size_t)j * OUT_N + i];
  out[i] = s * pgrads[i];
}

// ---------------------------------------------------------------------------
extern "C" void kernel_launch(void* const* d_in, const int* in_sizes, int n_in,
                              void* d_out, int out_size, void* d_ws,
                              size_t ws_size, hipStream_t stream) {
  const float* grad   = (const float*)d_in[0];  // (16,896,896,3) f32
  const int*   boxes  = (const int*)d_in[1];    // (16,16,4) i32
  /* d_in[2] transform_decisions: unused by the reference */
  const float* pgrads = (const float*)d_in[3];  // (64,64,3) f32
  float*       out    = (float*)d_out;          // (64,64,3) f32

  // Pick slice count that fits the workspace (constant per environment ->
  // deterministic). Fallback S=1 accumulates through d_out (no ws needed).
  int S = 32;
  while (S > 1 && (size_t)S * OUT_N * sizeof(float) > ws_size) S >>= 1;
  float* partial = (float*)d_ws;
  if ((size_t)S * OUT_N * sizeof(float) > ws_size || d_ws == nullptr) {
    S = 1;
    partial = out;
  }
  const int pps = NPATCH / S;

  dim3 g1(PIX / 256, S);
  gp_stage1<<<g1, 256, 0, stream>>>(grad, boxes, partial, pps);
  gp_stage2<<<(OUT_N + 255) / 256, 256, 0, stream>>>(partial, pgrads, out, S);
}